// LinearAttention_16793322127870
// MI455X (gfx1250) — compile-verified
//
#include <hip/hip_runtime.h>
#include <hip/hip_bf16.h>
#include <math.h>

// ---------------------------------------------------------------------------
// Linear attention on MI455X (gfx1250), all matmuls via V_WMMA_F32_16X16X4_F32
// (full fp32 precision — reference is fp32 with K=1024 reductions, so bf16/f16
// matrix ops would not be numerically faithful).
//
// Fragment layouts (CDNA5 ISA 7.12.2, wave32):
//   A 16x4 f32 : lanes 0-15 -> M=lane,   VGPR0=K0, VGPR1=K1
//                lanes 16-31-> M=lane-16,VGPR0=K2, VGPR1=K3
//   B 4x16 f32 : mirrored (lanes 0-15 rows K0/K1, lanes 16-31 rows K2/K3)
//   C 16x16 f32: VGPR i: lanes 0-15 -> M=i, lanes 16-31 -> M=i+8, N=lane%16
// ---------------------------------------------------------------------------

typedef __attribute__((ext_vector_type(2))) float v2f;
typedef __attribute__((ext_vector_type(8))) float v8f;

#define DIM 1024
#define NHEADS 16
#define HDIM 64
#define BATCH 4
#define SEQ 4096
#define ROWS (BATCH * SEQ)          // 16384
#define FM_ROWS (ROWS * NHEADS)     // 262144

struct GemmParams {
  const float* A;
  const float* B0; const float* B1; const float* B2;
  float* C0; float* C1; float* C2;
  const float* mask;   // mode 2
  const float* norm;   // mode 3
  int K, lda, ldb, ldc;
  int zdiv;            // blockIdx.z = zq*zdiv + zr decomposition for offsets
  long long aS1, aS2, bS1, bS2, cS1, cS2, nS1, nS2;
  int mode;            // 0 plain | 1 elu+1 | 2 (elu+1)*mask | 3 /norm
};

static __device__ __forceinline__ v8f wmma_f32(v2f a, v2f b, v8f c) {
  return __builtin_amdgcn_wmma_f32_16x16x4_f32(
      false, a, false, b, (short)0, c, false, false);
}

// Generic fp32 WMMA GEMM: 64x64 block tile, BK=32, 256 threads (8 waves).
// Wave w: M-subtile (w>>1)*16, N-half (w&1)*32 -> two 16x16 accumulators.
__global__ __launch_bounds__(256)
void gemm_wmma_f32_kernel(GemmParams p) {
  __shared__ float As[64][32];
  __shared__ float Bs[32][64];

  const int tid   = threadIdx.x;
  const int lane  = tid & 31;
  const int wave  = tid >> 5;
  const int lhalf = lane >> 4;      // 0: holds K{0,1}; 1: holds K{2,3}
  const int lmod  = lane & 15;
  const int wm    = wave >> 1;      // 0..3
  const int wn    = wave & 1;       // 0..1

  const int z  = blockIdx.z;
  const int zq = z / p.zdiv;
  const int zr = z % p.zdiv;

  const float* A = p.A + (long long)zq * p.aS1 + (long long)zr * p.aS2;
  const float* B = (z == 0) ? p.B0 : ((z == 1) ? p.B1 : p.B2);
  B += (long long)zq * p.bS1 + (long long)zr * p.bS2;
  float* C = (z == 0) ? p.C0 : ((z == 1) ? p.C1 : p.C2);
  C += (long long)zq * p.cS1 + (long long)zr * p.cS2;

  const int m0 = blockIdx.y * 64;
  const int n0 = blockIdx.x * 64;

  v8f acc[2] = {};

  for (int k0 = 0; k0 < p.K; k0 += 32) {
    // Stage A tile (64x32) — 2048 floats, 2 float4 per thread, coalesced.
#pragma unroll
    for (int i = 0; i < 2; ++i) {
      const int f4 = tid + i * 256;
      const int r = f4 >> 3;            // 8 float4 per 32-float row
      const int c = (f4 & 7) << 2;
      *(float4*)&As[r][c] =
          *(const float4*)(A + (long long)(m0 + r) * p.lda + (k0 + c));
    }
    // Stage B tile (32x64).
#pragma unroll
    for (int i = 0; i < 2; ++i) {
      const int f4 = tid + i * 256;
      const int r = f4 >> 4;            // 16 float4 per 64-float row
      const int c = (f4 & 15) << 2;
      *(float4*)&Bs[r][c] =
          *(const float4*)(B + (long long)(k0 + r) * p.ldb + (n0 + c));
    }
    // Prefetch next A chunk toward L2 while we compute on this one.
    if (k0 + 32 < p.K)
      __builtin_prefetch(A + (long long)(m0 + (tid >> 3)) * p.lda + k0 + 32, 0, 1);
    __syncthreads();

#pragma unroll
    for (int k4 = 0; k4 < 8; ++k4) {
      const int kb = k4 * 4 + lhalf * 2;
      v2f a;
      a.x = As[wm * 16 + lmod][kb];
      a.y = As[wm * 16 + lmod][kb + 1];
#pragma unroll
      for (int t = 0; t < 2; ++t) {
        const int n = wn * 32 + t * 16 + lmod;
        v2f b;
        b.x = Bs[kb][n];
        b.y = Bs[kb + 1][n];
        acc[t] = wmma_f32(a, b, acc[t]);
      }
    }
    __syncthreads();
  }

  // Epilogue + store.
#pragma unroll
  for (int t = 0; t < 2; ++t) {
    const int col = n0 + wn * 32 + t * 16 + lmod;
#pragma unroll
    for (int i = 0; i < 8; ++i) {
      const int r = m0 + wm * 16 + lhalf * 8 + i;
      float v = acc[t][i];
      if (p.mode == 1 || p.mode == 2) {
        v = (v > 0.0f) ? (v + 1.0f) : __expf(v);   // elu(x)+1
        if (p.mode == 2) v *= p.mask[r >> 4];      // row -> (b*n) index
      } else if (p.mode == 3) {
        v /= p.norm[(long long)zq * p.nS1 + (long long)zr * p.nS2 + r];
      }
      C[(long long)r * p.ldc + col] = v;
    }
  }
}

// Per-(b,h): kv[64][64] = phiK^T @ V over n=4096, fused ksum (column sums of
// phiK). 4 waves, wave w owns kv rows [w*16, w*16+16), all 64 cols.
__global__ __launch_bounds__(128)
void kv_ksum_kernel(const float* __restrict__ phiK, const float* __restrict__ V,
                    float* __restrict__ kv, float* __restrict__ ksum) {
  __shared__ float Ks[64][64];
  __shared__ float Vs[64][64];
  __shared__ float red[4][32];

  const int z = blockIdx.x;         // b*16 + h
  const int b = z >> 4, h = z & 15;
  const int tid   = threadIdx.x;
  const int lane  = tid & 31;
  const int wave  = tid >> 5;       // 0..3 -> kv row tile
  const int lhalf = lane >> 4;
  const int lmod  = lane & 15;

  const long long base = ((long long)b * SEQ) * DIM + h * HDIM;

  v8f acc[4] = {};
  float ks = 0.0f;

  for (int n0 = 0; n0 < SEQ; n0 += 64) {
#pragma unroll
    for (int i = 0; i < 8; ++i) {
      const int f4 = tid + i * 128;  // 1024 float4 = 4096 floats per buffer
      const int r = f4 >> 4;
      const int c = (f4 & 15) << 2;
      *(float4*)&Ks[r][c] = *(const float4*)(phiK + base + (long long)(n0 + r) * DIM + c);
      *(float4*)&Vs[r][c] = *(const float4*)(V    + base + (long long)(n0 + r) * DIM + c);
    }
    __syncthreads();
#pragma unroll
    for (int k4 = 0; k4 < 16; ++k4) {
      const int kb = k4 * 4 + lhalf * 2;
      v2f a;                               // A = phiK^T : A[m][k] = Ks[k][m]
      a.x = Ks[kb][wave * 16 + lmod];
      a.y = Ks[kb + 1][wave * 16 + lmod];
      ks += a.x + a.y;                     // fused column-sum of phiK
#pragma unroll
      for (int t = 0; t < 4; ++t) {
        v2f bb;
        bb.x = Vs[kb][t * 16 + lmod];
        bb.y = Vs[kb + 1][t * 16 + lmod];
        acc[t] = wmma_f32(a, bb, acc[t]);
      }
    }
    __syncthreads();
  }

  float* kvz = kv + (long long)z * (HDIM * HDIM);
#pragma unroll
  for (int t = 0; t < 4; ++t) {
    const int col = t * 16 + lmod;
#pragma unroll
    for (int i = 0; i < 8; ++i) {
      const int r = wave * 16 + lhalf * 8 + i;
      kvz[r * HDIM + col] = acc[t][i];
    }
  }

  red[wave][lane] = ks;
  __syncthreads();
  if (lane < 16)
    ksum[z * HDIM + wave * 16 + lmod] = red[wave][lmod] + red[wave][lmod + 16];
}

// normalizer[z*4096 + n] = dot(phiQ[b,n,h,:], ksum[z,:]) + 1e-6
__global__ __launch_bounds__(256)
void norm_kernel(const float* __restrict__ phiQ, const float* __restrict__ ksum,
                 float* __restrict__ norm) {
  const int idx = blockIdx.x * 256 + threadIdx.x;  // z*4096 + n
  const int z = idx >> 12;
  const int n = idx & 4095;
  const int b = z >> 4, h = z & 15;
  const float* q = phiQ + ((long long)(b * SEQ + n)) * DIM + h * HDIM;
  const float* s = ksum + z * HDIM;
  float acc = 0.0f;
#pragma unroll
  for (int d = 0; d < HDIM; ++d) acc += q[d] * s[d];
  norm[idx] = acc + 1e-6f;
}

extern "C" void kernel_launch(void* const* d_in, const int* in_sizes, int n_in,
                              void* d_out, int out_size, void* d_ws, size_t ws_size,
                              hipStream_t stream) {
  const float* x    = (const float*)d_in[0];
  const float* mask = (const float*)d_in[1];
  const float* Wq   = (const float*)d_in[2];
  const float* Wk   = (const float*)d_in[3];
  const float* Wv   = (const float*)d_in[4];
  const float* Wo   = (const float*)d_in[5];
  const float* proj = (const float*)d_in[6];
  float* out = (float*)d_out;

  float* ws = (float*)d_ws;
  float* Q    = ws;                                   // 16384x1024
  float* Kb   = Q  + (long long)ROWS * DIM;           // 16384x1024
  float* Vb   = Kb + (long long)ROWS * DIM;           // 16384x1024
  float* kv   = Vb + (long long)ROWS * DIM;           // 64 x 64x64
  float* ksum = kv + (long long)BATCH * NHEADS * HDIM * HDIM;  // 64x64
  float* nrm  = ksum + (long long)BATCH * NHEADS * HDIM;       // 64x4096

  // 1) Fused Q/K/V projection: (16384x1024)@(1024x1024), z selects W/C.
  {
    GemmParams p = {};
    p.A = x; p.B0 = Wq; p.B1 = Wk; p.B2 = Wv;
    p.C0 = Q; p.C1 = Kb; p.C2 = Vb;
    p.K = DIM; p.lda = DIM; p.ldb = DIM; p.ldc = DIM;
    p.zdiv = 1; p.mode = 0;
    hipLaunchKernelGGL(gemm_wmma_f32_kernel, dim3(DIM / 64, ROWS / 64, 3),
                       dim3(256), 0, stream, p);
  }
  // 2) Feature map on Q (in-place): (262144x64)@proj(64x64), elu+1.
  {
    GemmParams p = {};
    p.A = Q; p.B0 = proj; p.B1 = proj; p.B2 = proj;
    p.C0 = Q; p.C1 = Q; p.C2 = Q;
    p.K = HDIM; p.lda = HDIM; p.ldb = HDIM; p.ldc = HDIM;
    p.zdiv = 1; p.mode = 1;
    hipLaunchKernelGGL(gemm_wmma_f32_kernel, dim3(1, FM_ROWS / 64, 1),
                       dim3(256), 0, stream, p);
  }
  // 3) Feature map on K (in-place) + mask.
  {
    GemmParams p = {};
    p.A = Kb; p.B0 = proj; p.B1 = proj; p.B2 = proj;
    p.C0 = Kb; p.C1 = Kb; p.C2 = Kb;
    p.mask = mask;
    p.K = HDIM; p.lda = HDIM; p.ldb = HDIM; p.ldc = HDIM;
    p.zdiv = 1; p.mode = 2;
    hipLaunchKernelGGL(gemm_wmma_f32_kernel, dim3(1, FM_ROWS / 64, 1),
                       dim3(256), 0, stream, p);
  }
  // 4) kv = phiK^T V and ksum, per (b,h).
  hipLaunchKernelGGL(kv_ksum_kernel, dim3(BATCH * NHEADS), dim3(128), 0, stream,
                     Kb, Vb, kv, ksum);
  // 5) normalizer.
  hipLaunchKernelGGL(norm_kernel, dim3(FM_ROWS / 256), dim3(256), 0, stream,
                     Q, ksum, nrm);
  // 6) qkv = phiQ @ kv, divided by normalizer, per (b,h), in-place over Q.
  {
    GemmParams p = {};
    p.A = Q; p.B0 = kv; p.B1 = kv; p.B2 = kv;
    p.C0 = Q; p.C1 = Q; p.C2 = Q;
    p.norm = nrm;
    p.K = HDIM; p.lda = DIM; p.ldb = HDIM; p.ldc = DIM;
    p.zdiv = NHEADS;
    p.aS1 = (long long)SEQ * DIM; p.aS2 = HDIM;            // z = b*16+h
    p.bS1 = (long long)NHEADS * HDIM * HDIM; p.bS2 = HDIM * HDIM;
    p.cS1 = (long long)SEQ * DIM; p.cS2 = HDIM;
    p.nS1 = (long long)NHEADS * SEQ; p.nS2 = SEQ;
    p.mode = 3;
    hipLaunchKernelGGL(gemm_wmma_f32_kernel, dim3(1, SEQ / 64, BATCH * NHEADS),
                       dim3(256), 0, stream, p);
  }
  // 7) Output projection: (16384x1024)@Wo(1024x1024) -> d_out.
  {
    GemmParams p = {};
    p.A = Q; p.B0 = Wo; p.B1 = Wo; p.B2 = Wo;
    p.C0 = out; p.C1 = out; p.C2 = out;
    p.K = DIM; p.lda = DIM; p.ldb = DIM; p.ldc = DIM;
    p.zdiv = 1; p.mode = 0;
    hipLaunchKernelGGL(gemm_wmma_f32_kernel, dim3(DIM / 64, ROWS / 64, 1),
                       dim3(256), 0, stream, p);
  }
  (void)in_sizes; (void)n_in; (void)out_size; (void)ws_size;
}